// SODAugNN_17583596110490
// MI455X (gfx1250) — compile-verified
//
#include <hip/hip_runtime.h>

#define N_NODES 20000
#define D_FEAT  256
#define TWO_D   512
#define N_EDGES 320000
#define M_TILES (N_NODES / 16)   // 1250, exact

typedef __attribute__((ext_vector_type(2))) float v2f;
typedef __attribute__((ext_vector_type(8))) float v8f;

__device__ __forceinline__ void atomAddF(float* p, float v) {
    // lowers to global_atomic_add_f32 (no CAS loop)
    unsafeAtomicAdd(p, v);
}

// ---------------------------------------------------------------------------
// Edge phase: agg[dst] += relu(x[src] + edge_attr[e]); one thread = 4 floats.
// ---------------------------------------------------------------------------
__global__ void __launch_bounds__(256) edge_msg_kernel(
    const float* __restrict__ x, const long long* __restrict__ ei,
    const float* __restrict__ ea, float* __restrict__ agg) {
    int t  = blockIdx.x * 256 + threadIdx.x;           // E * (D/4) = 20.48M
    int e  = t >> 6;                                   // D/4 = 64 vec4 per edge
    int d4 = (t & 63) << 2;
    if (e >= N_EDGES) return;
    int src = (int)ei[e];
    int dst = (int)ei[N_EDGES + e];
    const float4 xv = *reinterpret_cast<const float4*>(x  + (size_t)src * D_FEAT + d4);
    const float4 av = *reinterpret_cast<const float4*>(ea + (size_t)e   * D_FEAT + d4);
    float* p = agg + (size_t)dst * D_FEAT + d4;
    atomAddF(p + 0, fmaxf(xv.x + av.x, 0.f));
    atomAddF(p + 1, fmaxf(xv.y + av.y, 0.f));
    atomAddF(p + 2, fmaxf(xv.z + av.z, 0.f));
    atomAddF(p + 3, fmaxf(xv.w + av.w, 0.f));
}

// ---------------------------------------------------------------------------
// WMMA GEMM: Out[M,NCOLS] = f(A) @ W[K,NCOLS] + bias
//   MODE 0: f(A) = A + A2           (x + agg), also accumulates BN col stats
//   MODE 1: f(A) = relu?(A*scale[k] + shift[k])   (BN + optional ReLU)
// One wave -> 16x64 strip (4x v8f acc). 8 waves/block -> 64 rows x 128 cols.
// All per-wave predicates are wave-uniform => EXEC all-ones for every WMMA.
// ---------------------------------------------------------------------------
template<int K, int NCOLS, int MODE, bool RELU>
__global__ void __launch_bounds__(256) gemm_wmma_kernel(
    const float* __restrict__ A, const float* __restrict__ A2,
    const float* __restrict__ W, const float* __restrict__ bias,
    const float* __restrict__ scale, const float* __restrict__ shift,
    float* __restrict__ Out, float* __restrict__ colsum, float* __restrict__ colsq) {
    const int lane = threadIdx.x & 31;
    const int wave = threadIdx.x >> 5;
    const int mt   = blockIdx.x * 4 + (wave & 3);      // 16-row tile index
    if (mt >= M_TILES) return;                          // wave-uniform exit
    const int n0    = blockIdx.y * 128 + (wave >> 2) * 64;
    const int row   = mt * 16 + (lane & 15);
    const int khalf = (lane >> 4) * 2;                  // 0 | 2 (A/B lane split)

    v8f acc0 = {0,0,0,0,0,0,0,0};
    v8f acc1 = acc0, acc2 = acc0, acc3 = acc0;

    const size_t arow = (size_t)row * K;
    for (int k = 0; k < K; k += 4) {
        const int kk = k + khalf;
        v2f a;
        if (MODE == 0) {
            a.x = A[arow + kk]     + A2[arow + kk];
            a.y = A[arow + kk + 1] + A2[arow + kk + 1];
        } else {
            float h0 = A[arow + kk]     * scale[kk]     + shift[kk];
            float h1 = A[arow + kk + 1] * scale[kk + 1] + shift[kk + 1];
            if (RELU) { h0 = fmaxf(h0, 0.f); h1 = fmaxf(h1, 0.f); }
            a.x = h0; a.y = h1;
        }
        const float* wrow0 = W + (size_t)kk * NCOLS + n0 + (lane & 15);
        const float* wrow1 = wrow0 + NCOLS;
        v2f b;
        b.x = wrow0[0];  b.y = wrow1[0];
        acc0 = __builtin_amdgcn_wmma_f32_16x16x4_f32(false, a, false, b, (short)0, acc0, false, false);
        b.x = wrow0[16]; b.y = wrow1[16];
        acc1 = __builtin_amdgcn_wmma_f32_16x16x4_f32(false, a, false, b, (short)0, acc1, false, false);
        b.x = wrow0[32]; b.y = wrow1[32];
        acc2 = __builtin_amdgcn_wmma_f32_16x16x4_f32(false, a, false, b, (short)0, acc2, false, false);
        b.x = wrow0[48]; b.y = wrow1[48];
        acc3 = __builtin_amdgcn_wmma_f32_16x16x4_f32(false, a, false, b, (short)0, acc3, false, false);
    }

    // Epilogue: +bias, store, optional BN column partial sums.
    const int mbase = mt * 16 + (lane >> 4) * 8;        // C/D: M = vgpr + 8*(lane>=16)
    v8f accs[4] = {acc0, acc1, acc2, acc3};
    #pragma unroll
    for (int s = 0; s < 4; ++s) {
        const int col = n0 + s * 16 + (lane & 15);
        const float bc = bias[col];
        float psum = 0.f, psq = 0.f;
        #pragma unroll
        for (int r = 0; r < 8; ++r) {
            float v = accs[s][r] + bc;
            Out[(size_t)(mbase + r) * NCOLS + col] = v;
            psum += v; psq += v * v;
        }
        if (MODE == 0) {
            atomAddF(&colsum[col], psum);
            atomAddF(&colsq[col],  psq);
        }
    }
}

// ---------------------------------------------------------------------------
// BN stats -> per-column affine: scale = g*rsqrt(var+eps), shift = be - mu*scale
// ---------------------------------------------------------------------------
__global__ void __launch_bounds__(256) bn_finalize_kernel(
    const float* __restrict__ colsum, const float* __restrict__ colsq,
    const float* __restrict__ g, const float* __restrict__ be,
    float* __restrict__ scale, float* __restrict__ shift) {
    int n = blockIdx.x * 256 + threadIdx.x;
    if (n >= TWO_D) return;
    const float invM = 1.0f / (float)N_NODES;
    float mu  = colsum[n] * invM;
    float var = colsq[n] * invM - mu * mu;
    float sc  = g[n] * rsqrtf(var + 1e-5f);
    scale[n] = sc;
    shift[n] = be[n] - mu * sc;
}

// ---------------------------------------------------------------------------
// z = noise*exp(logstd) + mean;  kl = 0.5/N^2 * sum(1 + 2*ls - m^2 - exp(ls)^2)
// ---------------------------------------------------------------------------
__global__ void __launch_bounds__(256) finalize_kernel(
    const float* __restrict__ mean, const float* __restrict__ logstd,
    const float* __restrict__ noise, float* __restrict__ z, float* __restrict__ kl) {
    __shared__ float red[8];
    int t = blockIdx.x * 256 + threadIdx.x;
    float term = 0.f;
    if (t < N_NODES * D_FEAT) {
        float m = mean[t], ls = logstd[t];
        float s = expf(ls);
        z[t] = noise[t] * s + m;
        term = 1.f + 2.f * ls - m * m - s * s;
    }
    #pragma unroll
    for (int off = 16; off > 0; off >>= 1) term += __shfl_down(term, off, 32);
    if ((threadIdx.x & 31) == 0) red[threadIdx.x >> 5] = term;
    __syncthreads();
    if (threadIdx.x == 0) {
        float v = 0.f;
        #pragma unroll
        for (int i = 0; i < 8; ++i) v += red[i];
        atomAddF(kl, v * (0.5f / ((float)N_NODES * (float)N_NODES)));
    }
}

// ---------------------------------------------------------------------------
extern "C" void kernel_launch(void* const* d_in, const int* in_sizes, int n_in,
                              void* d_out, int out_size, void* d_ws, size_t ws_size,
                              hipStream_t stream) {
    const float*     x     = (const float*)d_in[0];
    const long long* ei    = (const long long*)d_in[1];   // int64 in reference
    const float*     ea    = (const float*)d_in[2];
    const float*     noise = (const float*)d_in[3];

    const size_t ND = (size_t)N_NODES * D_FEAT;
    float* outz   = (float*)d_out;            // [N,D]
    float* outm   = outz + ND;                // mean
    float* outls  = outm + ND;                // logstd
    float* outkl  = outls + ND;               // scalar

    float* agg   = (float*)d_ws;              // N*D
    float* H1    = agg + ND;                  // N*2D
    float* hb0   = H1 + 2 * ND;               // N*D
    float* hb1   = hb0 + ND;                  // N*D
    float* stats = hb1 + ND;                  // colsum[512] colsq[512] scale[512] shift[512]
    float* colsum = stats;
    float* colsq  = stats + TWO_D;
    float* scale  = stats + 2 * TWO_D;
    float* shift  = stats + 3 * TWO_D;

    const dim3 blk(256);
    const dim3 grdEdge((N_EDGES * (D_FEAT / 4) + 255) / 256);
    const dim3 grdG1((M_TILES + 3) / 4, TWO_D / 128);     // (313, 4)
    const dim3 grdG2((M_TILES + 3) / 4, D_FEAT / 128);    // (313, 2)
    const dim3 grdFin((int)((ND + 255) / 256));

    auto layer = [&](const float* xin, int i, float* out, bool relu) {
        const float* W1 = (const float*)d_in[4 + 6 * i];
        const float* b1 = (const float*)d_in[5 + 6 * i];
        const float* g  = (const float*)d_in[6 + 6 * i];
        const float* be = (const float*)d_in[7 + 6 * i];
        const float* W2 = (const float*)d_in[8 + 6 * i];
        const float* b2 = (const float*)d_in[9 + 6 * i];

        hipMemsetAsync(agg, 0, ND * sizeof(float), stream);
        hipMemsetAsync(stats, 0, 2 * TWO_D * sizeof(float), stream);
        edge_msg_kernel<<<grdEdge, blk, 0, stream>>>(xin, ei, ea, agg);
        gemm_wmma_kernel<D_FEAT, TWO_D, 0, false><<<grdG1, blk, 0, stream>>>(
            xin, agg, W1, b1, nullptr, nullptr, H1, colsum, colsq);
        bn_finalize_kernel<<<2, blk, 0, stream>>>(colsum, colsq, g, be, scale, shift);
        if (relu)
            gemm_wmma_kernel<TWO_D, D_FEAT, 1, true><<<grdG2, blk, 0, stream>>>(
                H1, nullptr, W2, b2, scale, shift, out, nullptr, nullptr);
        else
            gemm_wmma_kernel<TWO_D, D_FEAT, 1, false><<<grdG2, blk, 0, stream>>>(
                H1, nullptr, W2, b2, scale, shift, out, nullptr, nullptr);
    };

    layer(x,   0, hb0,   true);
    layer(hb0, 1, hb1,   true);
    layer(hb1, 2, outm,  false);
    layer(hb1, 3, outls, false);

    hipMemsetAsync(outkl, 0, sizeof(float), stream);
    finalize_kernel<<<grdFin, blk, 0, stream>>>(outm, outls, noise, outz, outkl);
}